// FairINV_53171695124560
// MI455X (gfx1250) — compile-verified
//
#include <hip/hip_runtime.h>
#include <hip/hip_bf16.h>

// GCN forward for MI455X (gfx1250, wave32).
// Dense GEMMs use V_WMMA_F32_16X16X4_F32 (exact f32 math, matches reference).
// Edge aggregation uses float4 gathers + HW f32 global atomics (bandwidth-bound path).

typedef float v2f __attribute__((ext_vector_type(2)));
typedef float v8f __attribute__((ext_vector_type(8)));

#define HID 128  // hidden dim fixed by the model; enables float4/tile vectorization

// ---------------------------------------------------------------- utilities
__global__ __launch_bounds__(256) void zero_f32(float* __restrict__ p, long n) {
  long i = (long)blockIdx.x * blockDim.x + threadIdx.x;
  long stride = (long)gridDim.x * blockDim.x;
  for (; i < n; i += stride) p[i] = 0.0f;
}

__global__ __launch_bounds__(256) void deg_kernel(const long long* __restrict__ ei,
                                                  const float* __restrict__ ew,
                                                  float* __restrict__ deg, int E) {
  int e = blockIdx.x * blockDim.x + threadIdx.x;
  if (e >= E) return;
  long long dst = ei[(long)E + e];
  unsafeAtomicAdd(&deg[dst], ew[e]);
}

__global__ __launch_bounds__(256) void dinv_kernel(const float* __restrict__ deg,
                                                   float* __restrict__ dinv,
                                                   float* __restrict__ dinv2, int n) {
  int i = blockIdx.x * blockDim.x + threadIdx.x;
  if (i >= n) return;
  float d = rsqrtf(deg[i] + 1.0f);  // self-loop weight 1.0
  dinv[i] = d;
  dinv2[i] = d * d;
}

// ---------------------------------------------------------------- dense GEMM
// Y[n_rows x 128] = X[n_rows x K] @ W[K x 128], f32 WMMA 16x16x4.
// One wave per 16-row M-tile; A fragment reused across the 8 column tiles.
__global__ __launch_bounds__(128) void gemm_wmma_f32(const float* __restrict__ X,
                                                     const float* __restrict__ W,
                                                     float* __restrict__ Y,
                                                     int n_rows, int K) {
  const int lane  = threadIdx.x & 31;
  const int wave  = threadIdx.x >> 5;
  const int mtile = blockIdx.x * 4 + wave;
  if (mtile * 16 >= n_rows) return;  // wave-uniform: EXEC stays all-1s for WMMA

  const int l15   = lane & 15;
  const int khalf = (lane >> 4) << 1;  // lanes 0-15 -> K+{0,1}; lanes 16-31 -> K+{2,3}
  const float* xrow = X + (size_t)(mtile * 16 + l15) * K;

  v8f acc[8] = {};
  for (int k0 = 0; k0 < K; k0 += 4) {
    float2 av = *(const float2*)(xrow + k0 + khalf);  // b64 load of A pair
    v2f a; a.x = av.x; a.y = av.y;
    const float* wp = W + (size_t)(k0 + khalf) * HID + l15;
#pragma unroll
    for (int t = 0; t < 8; ++t) {
      v2f b;
      b.x = wp[t * 16];
      b.y = wp[t * 16 + HID];
      acc[t] = __builtin_amdgcn_wmma_f32_16x16x4_f32(
          /*neg_a=*/false, a, /*neg_b=*/false, b,
          /*c_mod=*/(short)0, acc[t], /*reuse_a=*/false, /*reuse_b=*/false);
    }
  }

  // C/D layout: VGPR v, lanes 0-15 -> M=v, lanes 16-31 -> M=v+8; N = lane&15.
  const int mbase = mtile * 16 + ((lane >> 4) << 3);
#pragma unroll
  for (int t = 0; t < 8; ++t) {
    float* yp = Y + (size_t)mbase * HID + t * 16 + l15;
#pragma unroll
    for (int v = 0; v < 8; ++v) yp[(size_t)v * HID] = acc[t][v];
  }
}

// ---------------------------------------------------------------- edge scatter
// agg[dst] += H[src] * (dinv[src]*ew*dinv[dst]); one wave per edge, float4 lanes.
__global__ __launch_bounds__(256) void scatter_kernel(const float* __restrict__ H,
                                                      const long long* __restrict__ ei,
                                                      const float* __restrict__ ew,
                                                      const float* __restrict__ dinv,
                                                      float* __restrict__ agg, int E) {
  int e = blockIdx.x * 8 + (threadIdx.x >> 5);
  if (e >= E) return;
  int lane = threadIdx.x & 31;
  long long s = ei[e];
  long long d = ei[(long)E + e];
  float norm = dinv[s] * ew[e] * dinv[d];
  float4 hv = ((const float4*)(H + (size_t)s * HID))[lane];
  float* ap = agg + (size_t)d * HID + (lane << 2);
  unsafeAtomicAdd(ap + 0, hv.x * norm);
  unsafeAtomicAdd(ap + 1, hv.y * norm);
  unsafeAtomicAdd(ap + 2, hv.z * norm);
  unsafeAtomicAdd(ap + 3, hv.w * norm);
}

// ---------------------------------------------------------------- combine
// agg[n,f] += Hpre[n,f] * dinv2[n] + b[f]   (in place -> layer output)
__global__ __launch_bounds__(256) void combine_kernel(float* __restrict__ agg,
                                                      const float* __restrict__ Hpre,
                                                      const float* __restrict__ dinv2,
                                                      const float* __restrict__ b,
                                                      long total) {
  long i = (long)blockIdx.x * blockDim.x + threadIdx.x;
  if (i >= total) return;
  int f = (int)(i & (HID - 1));
  long n = i >> 7;
  agg[i] += Hpre[i] * dinv2[n] + b[f];
}

// ---------------------------------------------------------------- fused head
// h2 = agg2 + h2_pre*dinv2 + b2 ;  out[n] = dot(h2[n,:], Wc) + bc
__global__ __launch_bounds__(256) void head_kernel(const float* __restrict__ agg2,
                                                   const float* __restrict__ Hpre,
                                                   const float* __restrict__ dinv2,
                                                   const float* __restrict__ b2,
                                                   const float* __restrict__ Wc,
                                                   const float* __restrict__ bc,
                                                   float* __restrict__ out, int n_nodes) {
  int n = blockIdx.x * 8 + (threadIdx.x >> 5);
  if (n >= n_nodes) return;
  int lane = threadIdx.x & 31;
  float4 a = ((const float4*)(agg2 + (size_t)n * HID))[lane];
  float4 h = ((const float4*)(Hpre + (size_t)n * HID))[lane];
  float4 bb = ((const float4*)b2)[lane];
  float4 w = ((const float4*)Wc)[lane];
  float d2 = dinv2[n];
  float s = (a.x + h.x * d2 + bb.x) * w.x
          + (a.y + h.y * d2 + bb.y) * w.y
          + (a.z + h.z * d2 + bb.z) * w.z
          + (a.w + h.w * d2 + bb.w) * w.w;
#pragma unroll
  for (int off = 16; off > 0; off >>= 1) s += __shfl_xor(s, off, 32);
  if (lane == 0) out[n] = s + bc[0];
}

// ---------------------------------------------------------------- launcher
extern "C" void kernel_launch(void* const* d_in, const int* in_sizes, int n_in,
                              void* d_out, int out_size, void* d_ws, size_t ws_size,
                              hipStream_t stream) {
  const float*     x  = (const float*)d_in[0];
  const long long* ei = (const long long*)d_in[1];  // int64 edge_index [2,E]
  const float*     ew = (const float*)d_in[2];
  const float*     W1 = (const float*)d_in[3];
  const float*     b1 = (const float*)d_in[4];
  const float*     W2 = (const float*)d_in[5];
  const float*     b2 = (const float*)d_in[6];
  const float*     Wc = (const float*)d_in[7];
  const float*     bc = (const float*)d_in[8];

  const int hid    = in_sizes[4];          // 128
  const int in_dim = in_sizes[3] / hid;    // 256
  const int Nn     = in_sizes[0] / in_dim; // 50000
  const int Ee     = in_sizes[2];          // 800000

  // workspace layout (~52 MB)
  float* bufA  = (float*)d_ws;                 // h_pre
  float* bufB  = bufA + (size_t)Nn * hid;      // agg / layer output
  float* deg   = bufB + (size_t)Nn * hid;
  float* dinv  = deg + Nn;
  float* dinv2 = dinv + Nn;

  const long nh = (long)Nn * hid;
  const int  mtiles = (Nn + 15) / 16;          // 3125 (exact)

  // normalization terms
  zero_f32<<<256, 256, 0, stream>>>(deg, Nn);
  deg_kernel<<<(Ee + 255) / 256, 256, 0, stream>>>(ei, ew, deg, Ee);
  dinv_kernel<<<(Nn + 255) / 256, 256, 0, stream>>>(deg, dinv, dinv2, Nn);

  // ---- layer 1: h1 = GCNConv(x, W1, b1)
  gemm_wmma_f32<<<(mtiles + 3) / 4, 128, 0, stream>>>(x, W1, bufA, Nn, in_dim);
  zero_f32<<<2048, 256, 0, stream>>>(bufB, nh);
  scatter_kernel<<<(Ee + 7) / 8, 256, 0, stream>>>(bufA, ei, ew, dinv, bufB, Ee);
  combine_kernel<<<(int)((nh + 255) / 256), 256, 0, stream>>>(bufB, bufA, dinv2, b1, nh);

  // ---- layer 2 + classifier head
  gemm_wmma_f32<<<(mtiles + 3) / 4, 128, 0, stream>>>(bufB, W2, bufA, Nn, hid);
  zero_f32<<<2048, 256, 0, stream>>>(bufB, nh);
  scatter_kernel<<<(Ee + 7) / 8, 256, 0, stream>>>(bufA, ei, ew, dinv, bufB, Ee);
  head_kernel<<<(Nn + 7) / 8, 256, 0, stream>>>(bufB, bufA, dinv2, b2, Wc, bc,
                                                (float*)d_out, Nn);
}